// DeformationtionNet_9474697855043
// MI455X (gfx1250) — compile-verified
//
#include <hip/hip_runtime.h>
#include <math.h>

#define NUM_JOINTS 24
#define NUM_VERTS  6890
#define BATCH      128
#define TILE_M     16
#define WAVES_PER_BLOCK 8
#define PI_F 3.14159265358979323846f

typedef __attribute__((ext_vector_type(2))) float v2f;
typedef __attribute__((ext_vector_type(8))) float v8f;

__constant__ int c_parents[NUM_JOINTS] = {
    -1, 0, 0, 0, 1, 2, 3, 4, 5, 6, 7, 8, 9, 9, 9, 12, 13, 14, 16, 17, 18, 19, 20, 21
};

// ---------------------------------------------------------------------------
// Kernel 1: per-batch joint transforms. One wave per batch.
// Produces Gc[b][24][16] (row-major 4x4 per joint) in workspace.
// ---------------------------------------------------------------------------
__global__ __launch_bounds__(32) void gc_kernel(const float* __restrict__ J,
                                                const float* __restrict__ pose,
                                                float* __restrict__ Gc) {
    __shared__ float sA[NUM_JOINTS][16];
    __shared__ float sG[NUM_JOINTS][16];
    const int b = blockIdx.x;
    const int m = threadIdx.x;

    if (m < NUM_JOINTS) {
        // Rodrigues, faithful to reference: theta = ||r + eps||, cos(theta*pi), sin(theta)
        const float* r = pose + ((size_t)b * NUM_JOINTS + m) * 3;
        float rx = r[0], ry = r[1], rz = r[2];
        float ex = rx + 1e-8f, ey = ry + 1e-8f, ez = rz + 1e-8f;
        float theta = sqrtf(ex * ex + ey * ey + ez * ez);
        float inv = 1.0f / theta;
        float hx = rx * inv, hy = ry * inv, hz = rz * inv;
        float c = cosf(theta * PI_F);
        float s = sinf(theta);
        float oc = 1.0f - c;

        float R00 = c + oc * hx * hx;
        float R01 = oc * hx * hy - s * hz;
        float R02 = oc * hx * hz + s * hy;
        float R10 = oc * hy * hx + s * hz;
        float R11 = c + oc * hy * hy;
        float R12 = oc * hy * hz - s * hx;
        float R20 = oc * hz * hx - s * hy;
        float R21 = oc * hz * hy + s * hx;
        float R22 = c + oc * hz * hz;

        const float* Jb = J + (size_t)b * NUM_JOINTS * 3;
        int p = c_parents[m];
        float tx, ty, tz;
        if (m == 0) {
            tx = Jb[0]; ty = Jb[1]; tz = Jb[2];
        } else {
            tx = Jb[m * 3 + 0] - Jb[p * 3 + 0];
            ty = Jb[m * 3 + 1] - Jb[p * 3 + 1];
            tz = Jb[m * 3 + 2] - Jb[p * 3 + 2];
        }
        sA[m][0] = R00; sA[m][1] = R01; sA[m][2]  = R02; sA[m][3]  = tx;
        sA[m][4] = R10; sA[m][5] = R11; sA[m][6]  = R12; sA[m][7]  = ty;
        sA[m][8] = R20; sA[m][9] = R21; sA[m][10] = R22; sA[m][11] = tz;
        sA[m][12] = 0.f; sA[m][13] = 0.f; sA[m][14] = 0.f; sA[m][15] = 1.f;
    }
    __syncthreads();

    if (m == 0) {
        // Serial kinematic chain: G[i] = G[parent[i]] @ A[i]
        for (int k = 0; k < 16; ++k) sG[0][k] = sA[0][k];
        for (int i = 1; i < NUM_JOINTS; ++i) {
            int p = c_parents[i];
            for (int r = 0; r < 4; ++r) {
                for (int cc = 0; cc < 4; ++cc) {
                    float acc = 0.f;
                    for (int k = 0; k < 4; ++k)
                        acc += sG[p][r * 4 + k] * sA[i][k * 4 + cc];
                    sG[i][r * 4 + cc] = acc;
                }
            }
        }
    }
    __syncthreads();

    if (m < NUM_JOINTS) {
        const float* Jb = J + (size_t)b * NUM_JOINTS * 3;
        float jx = Jb[m * 3 + 0], jy = Jb[m * 3 + 1], jz = Jb[m * 3 + 2];
        float* dst = Gc + ((size_t)b * NUM_JOINTS + m) * 16;
        for (int r = 0; r < 4; ++r) {
            float corr = sG[m][r * 4 + 0] * jx + sG[m][r * 4 + 1] * jy + sG[m][r * 4 + 2] * jz;
            dst[r * 4 + 0] = sG[m][r * 4 + 0];
            dst[r * 4 + 1] = sG[m][r * 4 + 1];
            dst[r * 4 + 2] = sG[m][r * 4 + 2];
            dst[r * 4 + 3] = sG[m][r * 4 + 3] - corr;
        }
    }
}

// ---------------------------------------------------------------------------
// Kernel 2: skinning. One wave per 16-vertex tile.
// C[16v x 16e] = sum over 6 k-chunks of W[16v x 4j] * Gc[4j x 16e] via
// v_wmma_f32_16x16x4_f32, then per-vertex Tv . Vh epilogue through LDS.
// All global loads are issued unconditionally (clamped addresses) and hoisted
// ahead of the WMMA chain so they pipeline under one s_wait_loadcnt.
// ---------------------------------------------------------------------------
__global__ __launch_bounds__(256) void skin_kernel(const float* __restrict__ V,
                                                   const float* __restrict__ W,
                                                   const float* __restrict__ Gc,
                                                   float* __restrict__ out) {
    // 16 rows x stride 20 floats: conflict-free writes (lane-major) and reads (row-major)
    __shared__ float lds_c[WAVES_PER_BLOCK][16 * 20];

    const int b    = blockIdx.y;
    const int lane = threadIdx.x & 31;
    const int wave = threadIdx.x >> 5;
    const int tile = blockIdx.x * WAVES_PER_BLOCK + wave;
    const int v0   = tile * TILE_M;
    if (v0 >= NUM_VERTS) return;  // wave-uniform

    const int n     = lane & 15;        // N index / vertex-in-tile for A
    const int khalf = (lane >> 4) << 1; // 0 for lanes 0-15, 2 for lanes 16-31

    const int  vv     = v0 + n;
    const bool valid  = (vv < NUM_VERTS);
    const int  vclamp = valid ? vv : (NUM_VERTS - 1);

    // ---- Issue all global loads up front (no EXEC juggling, one wait) ----
    // A fragments: W[vclamp][khalf + {0,1}] per k-chunk; 8B-aligned pairs -> b64
    const float* wrow = W + ((size_t)b * NUM_VERTS + vclamp) * NUM_JOINTS + khalf;
    v2f Araw[6];
#pragma unroll
    for (int kc = 0; kc < 6; ++kc) {
        Araw[kc] = *(const v2f*)(wrow + kc * 4);
    }

    // B fragments: 6 k-chunks of Gc (4x16 each). Lane layout mirrors A:
    // lanes 0-15 hold K rows {0,1}, lanes 16-31 hold K rows {2,3}.
    const float* gcb = Gc + (size_t)b * NUM_JOINTS * 16;
    v2f Bfrag[6];
#pragma unroll
    for (int kc = 0; kc < 6; ++kc) {
        int k = kc * 4 + khalf;
        Bfrag[kc].x = gcb[k * 16 + n];
        Bfrag[kc].y = gcb[(k + 1) * 16 + n];
    }

    // Epilogue vertex position (prefetched now to hide latency)
    const float* vp = V + ((size_t)b * NUM_VERTS + vclamp) * 3;
    float px = vp[0], py = vp[1], pz = vp[2];

    // Zero A for out-of-range vertices via per-lane select (no EXEC change)
    const v2f zero2 = {0.f, 0.f};
#pragma unroll
    for (int kc = 0; kc < 6; ++kc) {
        Araw[kc] = valid ? Araw[kc] : zero2;
    }

    // ---- WMMA accumulation chain: EXEC is all-ones here ----
    v8f C = {};
#pragma unroll
    for (int kc = 0; kc < 6; ++kc) {
        C = __builtin_amdgcn_wmma_f32_16x16x4_f32(
                /*neg_a=*/false, Araw[kc], /*neg_b=*/false, Bfrag[kc],
                /*c_mod=*/(short)0, C, /*reuse_a=*/false, /*reuse_b=*/false);
    }

    // Spill C to LDS: VGPR r holds row M=r (lanes 0-15) / M=r+8 (lanes 16-31), col N=lane&15
    float* ldsw = &lds_c[wave][0];
    const int rowoff = (lane >> 4) << 3;  // 0 or 8
#pragma unroll
    for (int r = 0; r < 8; ++r) {
        ldsw[(r + rowoff) * 20 + n] = C[r];
    }
    // Intra-wave LDS RAW fence (no barrier: waves diverge on tile bounds)
    asm volatile("s_wait_dscnt 0" ::: "memory");

    // Epilogue: lanes 0-15, one vertex each: out = Tv(3x4) . [V,1]
    if (lane < 16 && valid) {
        const float* c = ldsw + n * 20;
        float o0 = c[0] * px + c[1] * py + c[2]  * pz + c[3];
        float o1 = c[4] * px + c[5] * py + c[6]  * pz + c[7];
        float o2 = c[8] * px + c[9] * py + c[10] * pz + c[11];
        float* op = out + ((size_t)b * NUM_VERTS + vv) * 3;
        op[0] = o0;
        op[1] = o1;
        op[2] = o2;
    }
}

// ---------------------------------------------------------------------------
extern "C" void kernel_launch(void* const* d_in, const int* in_sizes, int n_in,
                              void* d_out, int out_size, void* d_ws, size_t ws_size,
                              hipStream_t stream) {
    (void)in_sizes; (void)n_in; (void)out_size; (void)ws_size;
    const float* V    = (const float*)d_in[0];
    const float* J    = (const float*)d_in[1];
    const float* pose = (const float*)d_in[2];
    const float* W    = (const float*)d_in[3];
    float* out = (float*)d_out;
    float* Gc  = (float*)d_ws;  // BATCH * 24 * 16 floats = 196608 B

    gc_kernel<<<BATCH, 32, 0, stream>>>(J, pose, Gc);

    const int tiles = (NUM_VERTS + TILE_M - 1) / TILE_M;                 // 431
    const int gx    = (tiles + WAVES_PER_BLOCK - 1) / WAVES_PER_BLOCK;   // 54
    skin_kernel<<<dim3(gx, BATCH), WAVES_PER_BLOCK * 32, 0, stream>>>(V, W, Gc, out);
}